// SeizureGNN_87548613362522
// MI455X (gfx1250) — compile-verified
//
#include <hip/hip_runtime.h>
#include <stdint.h>

typedef __attribute__((ext_vector_type(2))) float v2f;
typedef __attribute__((ext_vector_type(8))) float v8f;

// K0: deg[i] = 1.0 (self-loop contribution to degree)
__global__ void init_deg_kernel(float* __restrict__ deg, int n) {
    int i = blockIdx.x * blockDim.x + threadIdx.x;
    if (i < n) deg[i] = 1.0f;
}

// K1: deg[dst[e]] += 1 over all E edges
__global__ void degree_kernel(const long long* __restrict__ ei,
                              float* __restrict__ deg, int nE) {
    int e = blockIdx.x * blockDim.x + threadIdx.x;
    if (e < nE) {
        int d = (int)ei[(long long)nE + e];
        atomicAdd(&deg[d], 1.0f);
    }
}

// K2: dinv[i] = rsqrt(deg[i])  (deg >= 1 due to self loops)
__global__ void dinv_kernel(float* __restrict__ deg, int n) {
    int i = blockIdx.x * blockDim.x + threadIdx.x;
    if (i < n) deg[i] = rsqrtf(deg[i]);
}

// K3: layer-1 aggregation (F_in = 1 -> scalar message per edge)
__global__ void agg1_kernel(const long long* __restrict__ ei,
                            const float* __restrict__ x,
                            const float* __restrict__ dinv,
                            float* __restrict__ agg1, int nE) {
    int e = blockIdx.x * blockDim.x + threadIdx.x;
    if (e < nE) {
        int s = (int)ei[e];
        int d = (int)ei[(long long)nE + e];
        atomicAdd(&agg1[d], dinv[s] * dinv[d] * x[s]);
    }
}

// K4: h1[node,f] = relu((agg1[node] + dinv^2*x[node]) * W1[f] + b1[f])
__global__ void h1_kernel(const float* __restrict__ x,
                          const float* __restrict__ dinv,
                          const float* __restrict__ agg1,
                          const float* __restrict__ W1,
                          const float* __restrict__ b1,
                          float* __restrict__ h1, int n) {
    int t = blockIdx.x * blockDim.x + threadIdx.x;
    int node = t >> 5;
    int f = t & 31;
    if (node < n) {
        float di = dinv[node];
        float tot = agg1[node] + di * di * x[node];
        float v = tot * W1[f] + b1[f];
        h1[node * 32 + f] = v > 0.0f ? v : 0.0f;
    }
}

// K5: layer-2 aggregation, 32 lanes per edge (one feature per lane)
__global__ void agg2_kernel(const long long* __restrict__ ei,
                            const float* __restrict__ dinv,
                            const float* __restrict__ h1,
                            float* __restrict__ agg2, int nE) {
    int t = blockIdx.x * blockDim.x + threadIdx.x;   // covers E*32 = 102.4M
    int e = t >> 5;
    int f = t & 31;
    if (e < nE) {
        int s = (int)ei[e];
        int d = (int)ei[(long long)nE + e];
        float norm = dinv[s] * dinv[d];
        atomicAdd(&agg2[d * 32 + f], norm * h1[s * 32 + f]);
    }
}

// K6: h2 = relu((agg2 + dinv^2*h1) @ W2 + b2), column-summed over nodes,
//     via V_WMMA_F32_16X16X4_F32 (4 waves/block, 16 nodes/wave, 32 WMMAs/wave)
__global__ __launch_bounds__(128)
void h2_pool_kernel(const float* __restrict__ agg2,
                    const float* __restrict__ h1,
                    const float* __restrict__ dinv,
                    const float* __restrict__ W2,
                    const float* __restrict__ b2,
                    float* __restrict__ gsum, int n) {
    __shared__ float a2s[64 * 32];
    __shared__ float colsum[64];

    int tid = threadIdx.x;
    int blockBase = blockIdx.x * 64;

    if (tid < 64) colsum[tid] = 0.0f;

    for (int idx = tid; idx < 64 * 32; idx += 128) {
        int node = idx >> 5;
        int g = blockBase + node;
        float v = 0.0f;
        if (g < n) {
            float di = dinv[g];
            int off = g * 32 + (idx & 31);
            v = agg2[off] + di * di * h1[off];
        }
        a2s[idx] = v;
    }
    __syncthreads();

    int lane = tid & 31;
    int wave = tid >> 5;
    int row  = lane & 15;       // A: M within tile; B/D: N within tile
    int hi   = lane >> 4;       // selects K pair (A/B) and +8 row offset (D)
    const float* a2row = &a2s[(wave * 16 + row) * 32];

    for (int n0 = 0; n0 < 64; n0 += 16) {
        v8f c = {0.0f, 0.0f, 0.0f, 0.0f, 0.0f, 0.0f, 0.0f, 0.0f};
        int col = n0 + row;
        #pragma unroll
        for (int k0 = 0; k0 < 32; k0 += 4) {
            int ka = k0 + (hi ? 2 : 0);
            v2f a;                      // A 16x4 f32: lane half picks K0/K1 vs K2/K3
            a.x = a2row[ka];
            a.y = a2row[ka + 1];
            v2f b;                      // B 4x16 f32: row = vgpr + 2*hi
            b.x = W2[ka * 64 + col];
            b.y = W2[(ka + 1) * 64 + col];
            c = __builtin_amdgcn_wmma_f32_16x16x4_f32(
                    false, a, false, b, (short)0, c, false, false);
        }
        float bb = b2[col];
        float part = 0.0f;
        #pragma unroll
        for (int r = 0; r < 8; ++r) {   // D: M = r + 8*hi, N = col
            int node = blockBase + wave * 16 + r + 8 * hi;
            float v = c[r] + bb;
            if (node < n && v > 0.0f) part += v;
        }
        atomicAdd(&colsum[col], part);
    }
    __syncthreads();
    if (tid < 64) atomicAdd(&gsum[tid], colsum[tid]);
}

// K7: out[j] = sum_f (gsum[f]/N) * Wfc[f,j] + bfc[j]
__global__ void fc_kernel(const float* __restrict__ gsum,
                          const float* __restrict__ Wfc,
                          const float* __restrict__ bfc,
                          float* __restrict__ out, int n) {
    int j = threadIdx.x;
    if (j < 2) {
        float inv = 1.0f / (float)n;
        float acc = 0.0f;
        for (int f = 0; f < 64; ++f) acc += (gsum[f] * inv) * Wfc[f * 2 + j];
        out[j] = acc + bfc[j];
    }
}

extern "C" void kernel_launch(void* const* d_in, const int* in_sizes, int n_in,
                              void* d_out, int out_size, void* d_ws, size_t ws_size,
                              hipStream_t stream) {
    const float*     x   = (const float*)d_in[0];
    const long long* ei  = (const long long*)d_in[1];   // [2,E] int64
    const float*     W1  = (const float*)d_in[2];
    const float*     b1  = (const float*)d_in[3];
    const float*     W2  = (const float*)d_in[4];
    const float*     b2  = (const float*)d_in[5];
    const float*     Wfc = (const float*)d_in[6];
    const float*     bfc = (const float*)d_in[7];
    float*           out = (float*)d_out;

    const int N = in_sizes[0];        // 100000
    const int E = in_sizes[1] / 2;    // 3200000

    // ws layout (floats): deg/dinv[N] | agg1[N] | h1[N*32] | agg2[N*32] | gsum[64]
    float* ws   = (float*)d_ws;
    float* deg  = ws;
    float* agg1 = deg + N;
    float* h1   = agg1 + N;
    float* agg2 = h1 + (size_t)N * 32;
    float* gsum = agg2 + (size_t)N * 32;

    hipMemsetAsync(agg1, 0, sizeof(float) * (size_t)N, stream);
    hipMemsetAsync(agg2, 0, sizeof(float) * ((size_t)N * 32 + 64), stream);

    const int B = 256;
    init_deg_kernel<<<(N + B - 1) / B, B, 0, stream>>>(deg, N);
    degree_kernel<<<(E + B - 1) / B, B, 0, stream>>>(ei, deg, E);
    dinv_kernel<<<(N + B - 1) / B, B, 0, stream>>>(deg, N);
    agg1_kernel<<<(E + B - 1) / B, B, 0, stream>>>(ei, x, deg, agg1, E);
    h1_kernel<<<(N * 32 + B - 1) / B, B, 0, stream>>>(x, deg, agg1, W1, b1, h1, N);
    int agg2Threads = E * 32;         // 102.4M, fits in int
    agg2_kernel<<<(agg2Threads + B - 1) / B, B, 0, stream>>>(ei, deg, h1, agg2, E);
    h2_pool_kernel<<<(N + 63) / 64, 128, 0, stream>>>(agg2, h1, deg, W2, b2, gsum, N);
    fc_kernel<<<1, 64, 0, stream>>>(gsum, Wfc, bfc, out, N);
}